// PatchCovarianceNoiseEstimator_37417755082762
// MI455X (gfx1250) — compile-verified
//
#include <hip/hip_runtime.h>

// PatchCovarianceNoiseEstimator for MI455X (gfx1250, wave32).
// Stage 1: column sums of the implicit im2col matrix.
// Stage 2: Gram matrix G = P P^T via V_WMMA_F32_16X16X4_F32 (fp32 tensor path).
//          Branchless hot loop, unconditional loads, guarded tail K-step.
//          Each wave computes a 16x32 strip (2 col tiles share the A fragment).
// Stage 3: covariance -> parallel Jacobi eigensolver in 148KB LDS -> sort ->
//          prefix-mean threshold search -> sqrt(tau).

#define Himg 512
#define Wimg 512
#define Cch  3
#define NP   169                 // patches per spatial dim: (512-8)/3+1
#define NPATCH (NP * NP)         // 28561
#define Dd   192                 // C*p*p
#define Bb   8
#define LDA  193                 // LDS row stride (odd -> bank-friendly)

typedef float v2f __attribute__((ext_vector_type(2)));
typedef float v8f __attribute__((ext_vector_type(8)));

// ---------------- Stage 1: per-(b,d) patch sums ----------------
__global__ void col_sums_kernel(const float* __restrict__ x, float* __restrict__ sums) {
  int bd = blockIdx.x;                 // 0 .. B*D-1
  int b = bd / Dd, d = bd % Dd;
  int c = d >> 6, rem = d & 63, py = rem >> 3, px = rem & 7;
  const float* xb = x + (size_t)b * Cch * Himg * Wimg + c * Himg * Wimg + py * Wimg + px;
  float acc = 0.f;
  for (int n = threadIdx.x; n < NPATCH; n += blockDim.x) {
    int ny = n / NP, nx = n - ny * NP;
    acc += xb[ny * (3 * Wimg) + nx * 3];
  }
  __shared__ float red[256];
  red[threadIdx.x] = acc;
  __syncthreads();
  for (int off = 128; off > 0; off >>= 1) {
    if ((int)threadIdx.x < off) red[threadIdx.x] += red[threadIdx.x + off];
    __syncthreads();
  }
  if (threadIdx.x == 0) sums[bd] = red[0];
}

// ---------------- Stage 2: Gram via fp32 WMMA ----------------
// f32 A 16x4 layout (ISA 7.12.2): lanes 0-15 -> M=lane, K={0,1};
// lanes 16-31 -> M=lane-16, K={2,3}. B 4x16 mirrors.
__global__ void gram_wmma_kernel(const float* __restrict__ x, float* __restrict__ G) {
  int wave = blockIdx.x * (blockDim.x >> 5) + (threadIdx.x >> 5);
  int lane = threadIdx.x & 31;
  int b   = wave / 72;                 // 12 row-tiles * 6 col-pairs = 72 per batch
  int t   = wave % 72;
  int tr  = t / 6;
  int tcp = t % 6;                     // col pair
  const float* xb = x + (size_t)b * Cch * Himg * Wimg;

  // loop-invariant im2col offsets for this lane's row / two col d-indices
  int m   = lane & 15;
  int rd  = tr * 16 + m;
  int cd0 = (tcp * 2 + 0) * 16 + m;
  int cd1 = (tcp * 2 + 1) * 16 + m;
  int roff  = (rd  >> 6) * (Himg * Wimg) + ((rd  & 63) >> 3) * Wimg + (rd  & 7);
  int coff0 = (cd0 >> 6) * (Himg * Wimg) + ((cd0 & 63) >> 3) * Wimg + (cd0 & 7);
  int coff1 = (cd1 >> 6) * (Himg * Wimg) + ((cd1 & 63) >> 3) * Wimg + (cd1 & 7);

  int koff = (lane >> 4) << 1;         // 0 for lanes 0-15, 2 for lanes 16-31
  v8f acc0 = {};
  v8f acc1 = {};

  // per-lane patch walk: n0 = k0 + koff, stride 4; pix = ny*1536 + nx*3
  int nx   = koff;                     // column index of patch n0 (always < 169)
  int pix0 = koff * 3;

  // main loop: 7140 full K-steps cover n = 0..28559 (all in-bounds, no guards)
  for (int it = 0; it < 7140; ++it) {
    int pix1 = pix0 + ((nx == NP - 1) ? 1032 : 3);   // n0+1 (row wrap: +1536-504)
    v2f a, bv0, bv1;
    a.x   = xb[roff  + pix0];
    a.y   = xb[roff  + pix1];
    bv0.x = xb[coff0 + pix0];
    bv0.y = xb[coff0 + pix1];
    bv1.x = xb[coff1 + pix0];
    bv1.y = xb[coff1 + pix1];
    acc0 = __builtin_amdgcn_wmma_f32_16x16x4_f32(false, a, false, bv0,
                                                 (short)0, acc0, false, false);
    acc1 = __builtin_amdgcn_wmma_f32_16x16x4_f32(false, a, false, bv1,
                                                 (short)0, acc1, false, false);
    bool wr = (nx >= NP - 4);                        // stride-4 advance with wrap
    pix0 += wr ? 1041 : 12;                          // +1536 - 495
    nx    = wr ? nx - (NP - 4) : nx + 4;
  }

  // tail K-step: n = 28560..28563, only n == 28560 (koff==0, first slot) valid
  {
    bool tv = (koff == 0);
    v2f a, bv0, bv1;
    a.x   = tv ? xb[roff  + pix0] : 0.f;  a.y   = 0.f;
    bv0.x = tv ? xb[coff0 + pix0] : 0.f;  bv0.y = 0.f;
    bv1.x = tv ? xb[coff1 + pix0] : 0.f;  bv1.y = 0.f;
    acc0 = __builtin_amdgcn_wmma_f32_16x16x4_f32(false, a, false, bv0,
                                                 (short)0, acc0, false, false);
    acc1 = __builtin_amdgcn_wmma_f32_16x16x4_f32(false, a, false, bv1,
                                                 (short)0, acc1, false, false);
  }

  // C/D layout: VGPR r -> (M=r, N=lane) lanes 0-15; (M=r+8, N=lane-16) lanes 16-31
  int half = (lane >> 4) & 1;
  int col0 = (tcp * 2 + 0) * 16 + (lane & 15);
  int col1 = (tcp * 2 + 1) * 16 + (lane & 15);
  for (int r = 0; r < 8; ++r) {
    int row = tr * 16 + r + half * 8;
    G[((size_t)b * Dd + row) * Dd + col0] = acc0[r];
    G[((size_t)b * Dd + row) * Dd + col1] = acc1[r];
  }
}

// ---------------- Stage 3: Jacobi eigensolver + threshold ----------------
__global__ void eig_thresh_kernel(const float* __restrict__ G, const float* __restrict__ sums,
                                  float* __restrict__ out) {
  extern __shared__ float A[];         // Dd * LDA floats = 148224 B (CDNA5: 320KB LDS/WGP)
  __shared__ float cs_c[96], cs_s[96];
  __shared__ int   pr_p[96], pr_q[96];
  __shared__ float eig[Dd];
  __shared__ float csum[Dd];
  __shared__ float tauArr[Dd];
  __shared__ int   validArr[Dd];

  int b = blockIdx.x;
  int tid = threadIdx.x;
  const float invN = 1.0f / (float)NPATCH;

  // sigma = G/N - mu mu^T
  for (int idx = tid; idx < Dd * Dd; idx += blockDim.x) {
    int i = idx / Dd, j = idx % Dd;
    float mui = sums[b * Dd + i] * invN;
    float muj = sums[b * Dd + j] * invN;
    A[i * LDA + j] = G[((size_t)b * Dd + i) * Dd + j] * invN - mui * muj;
  }
  __syncthreads();

  const int mR = Dd - 1;               // 191 rounds per sweep (circle method)
  for (int sweep = 0; sweep < 10; ++sweep) {
    for (int r = 0; r < mR; ++r) {
      if (tid < Dd / 2) {
        int i = tid;
        int a1 = (i == 0) ? 0 : 1 + ((i - 1 + r) % mR);
        int j2 = (Dd - 1) - i;
        int a2 = 1 + ((j2 - 1 + r) % mR);
        int p = min(a1, a2), q = max(a1, a2);
        float app = A[p * LDA + p], aqq = A[q * LDA + q], apq = A[p * LDA + q];
        float c = 1.f, s = 0.f;
        if (fabsf(apq) > 1e-30f) {
          float tau = (aqq - app) / (2.f * apq);
          float tt  = (tau >= 0.f ? 1.f : -1.f) / (fabsf(tau) + sqrtf(1.f + tau * tau));
          c = rsqrtf(1.f + tt * tt);
          s = tt * c;
        }
        cs_c[i] = c; cs_s[i] = s; pr_p[i] = p; pr_q[i] = q;
      }
      __syncthreads();
      // rows: A <- J^T A   (96 disjoint pairs, conflict-free)
      for (int idx = tid; idx < (Dd / 2) * Dd; idx += blockDim.x) {
        int pi = idx / Dd, k = idx % Dd;
        int p = pr_p[pi], q = pr_q[pi];
        float c = cs_c[pi], s = cs_s[pi];
        float apk = A[p * LDA + k], aqk = A[q * LDA + k];
        A[p * LDA + k] = c * apk - s * aqk;
        A[q * LDA + k] = s * apk + c * aqk;
      }
      __syncthreads();
      // cols: A <- A J
      for (int idx = tid; idx < (Dd / 2) * Dd; idx += blockDim.x) {
        int pi = idx / Dd, k = idx % Dd;
        int p = pr_p[pi], q = pr_q[pi];
        float c = cs_c[pi], s = cs_s[pi];
        float akp = A[k * LDA + p], akq = A[k * LDA + q];
        A[k * LDA + p] = c * akp - s * akq;
        A[k * LDA + q] = s * akp + c * akq;
      }
      __syncthreads();
    }
  }

  if (tid < Dd) eig[tid] = A[tid * LDA + tid];
  __syncthreads();

  // odd-even transposition sort, ascending
  for (int phase = 0; phase < Dd; ++phase) {
    if (tid < Dd / 2) {
      int i = 2 * tid + (phase & 1);
      if (i + 1 < Dd) {
        float a0 = eig[i], a1v = eig[i + 1];
        if (a0 > a1v) { eig[i] = a1v; eig[i + 1] = a0; }
      }
    }
    __syncthreads();
  }

  if (tid == 0) {
    float s2 = 0.f;
    for (int i = 0; i < Dd; ++i) { s2 += eig[i]; csum[i] = s2; }
  }
  __syncthreads();

  // threshold search: k-th candidate uses prefix of length L = D-1-k
  if (tid < Dd) {
    int k = tid;
    int L = (Dd - 1) - k;
    float tau = (L > 0) ? csum[L - 1] / (float)L : 0.f;
    int ng = 0, nl = 0;
    for (int j = 0; j < L; ++j) {
      float e = eig[j];
      ng += (e > tau);
      nl += (e < tau);
    }
    tauArr[k]   = tau;
    validArr[k] = (ng == nl);
  }
  __syncthreads();

  if (tid == 0) {
    int idx = 0;
    for (int k = 0; k < Dd; ++k) { if (validArr[k]) { idx = k; break; } }
    out[b] = sqrtf(tauArr[idx]);
  }
}

extern "C" void kernel_launch(void* const* d_in, const int* in_sizes, int n_in,
                              void* d_out, int out_size, void* d_ws, size_t ws_size,
                              hipStream_t stream) {
  (void)in_sizes; (void)n_in; (void)out_size; (void)ws_size;
  const float* x = (const float*)d_in[0];
  float* out  = (float*)d_out;
  float* sums = (float*)d_ws;                 // B*D floats
  float* G    = sums + Bb * Dd;               // B*D*D floats (~1.18 MB)

  col_sums_kernel<<<Bb * Dd, 256, 0, stream>>>(x, sums);
  gram_wmma_kernel<<<72, 256, 0, stream>>>(x, G);      // 576 waves, 16x32 strip each
  size_t lds = (size_t)Dd * LDA * sizeof(float);       // 148224 B < 320 KB/WGP
  eig_thresh_kernel<<<Bb, 256, lds, stream>>>(G, sums, out);
}